// GNN_2138893713896
// MI455X (gfx1250) — compile-verified
//
#include <hip/hip_runtime.h>
#include <math.h>

// ---------------------------------------------------------------------------
// GAT GNN forward for MI455X (gfx1250, wave32).
// Dense GEMMs use V_WMMA_F32_16X16X4_F32 (full-fp32 WMMA path) with the
// 16xK A-tile staged in LDS (coalesced b128 global loads, conflict-free
// ds_load_b64 fragment reads via +4-dword row padding).
// Edge softmax/aggregation via float atomics (exact segment-max through a
// monotonic uint encoding). BatchNorm via two-pass column stats.
// ---------------------------------------------------------------------------

typedef __attribute__((ext_vector_type(2))) float v2f;
typedef __attribute__((ext_vector_type(4))) float v4f;
typedef __attribute__((ext_vector_type(8))) float v8f;

#define HIDC   128
#define HEADS  4
#define CH     32
#define NEG_SLOPE 0.2f
#define BN_EPS 1e-5f
#define KMAX   256
#define LDS_PAD 4                      // row stride = K + 4 dwords (bank spread)

__device__ __forceinline__ float elu1(float x)   { return x > 0.f ? x : expm1f(x); }
__device__ __forceinline__ float lrelu(float x)  { return x > 0.f ? x : NEG_SLOPE * x; }

// monotonic float <-> uint encoding: unsigned order == float order
__device__ __forceinline__ unsigned fenc(float f) {
    unsigned b = __float_as_uint(f);
    return (b & 0x80000000u) ? ~b : (b | 0x80000000u);
}
__device__ __forceinline__ float fdec(unsigned u) {
    unsigned b = (u & 0x80000000u) ? (u & 0x7fffffffu) : ~u;
    return __uint_as_float(b);
}

// src/dst of logical edge e (self loops appended after the E real edges)
__device__ __forceinline__ void edge_sd(int e, const int* __restrict__ ei, int E,
                                        int& s, int& d) {
    if (e < E) { s = ei[e]; d = ei[E + e]; }
    else       { s = e - E; d = e - E; }
}

// ---------------------------------------------------------------------------
// C[M x ncols] = act( A[M x K] @ B[K x ncols] + bias )
// A row stride is 128; if K == 256 the upper 128 K-columns come from A2
// (also row stride 128)  ->  implements concat([h, x]) @ W_skip.
// One wave per 16x16 output tile; blockDim.x = 32 * (ncols/16).
// The 16xK A-tile is staged in LDS once per block.
// WMMA f32 16x16x4 operand layout:
//   A (16x4): lanes 0-15 rows M, v0/v1 = K {0,1}; lanes 16-31 = K {2,3}
//   B (4x16): lanes 0-15 cols N, v0/v1 = K {0,1}; lanes 16-31 = K {2,3}
//   D (16x16): acc[v], lanes 0-15 -> (M=v, N=lane); 16-31 -> (M=v+8, N=lane-16)
// ---------------------------------------------------------------------------
__global__ void wmma_gemm_kernel(const float* __restrict__ A,
                                 const float* __restrict__ A2,
                                 const float* __restrict__ B,
                                 const float* __restrict__ bias,
                                 float* __restrict__ C,
                                 int M, int K, int ncols, int act)
{
    __shared__ float smem[16 * (KMAX + LDS_PAD)];

    const int wave = threadIdx.x >> 5;
    const int lane = threadIdx.x & 31;
    const int hi   = lane >> 4;        // half-wave: 0 -> K {0,1}, 1 -> K {2,3}
    const int mr   = lane & 15;
    const int m0   = blockIdx.x * 16;
    const int n0   = wave * 16;
    const int strideL = K + LDS_PAD;

    // --- cooperative, coalesced stage of the 16xK A-tile into LDS ---------
    for (int idx = threadIdx.x; idx < (16 * K) / 4; idx += blockDim.x) {
        const int el  = idx * 4;
        const int row = el / K;
        const int col = el % K;
        int r = m0 + row; if (r >= M) r = M - 1;   // clamp (stores guarded later)
        const float* src = (col < 128) ? (A  + (size_t)r * 128 + col)
                                       : (A2 + (size_t)r * 128 + (col - 128));
        *(v4f*)(&smem[row * strideL + col]) = *(const v4f*)src;  // ds_store_b128
    }
    __syncthreads();

    // --- K loop: LDS A fragments (bank-conflict free), global B fragments --
    v8f acc = {};
    #pragma unroll 4
    for (int kb = 0; kb < K; kb += 4) {
        v2f a = *(const v2f*)(&smem[mr * strideL + kb + 2 * hi]);  // ds_load_b64
        const float* bp = B + (size_t)(kb + 2 * hi) * ncols + n0 + mr;
        v2f b;
        b.x = bp[0];
        b.y = bp[ncols];
        acc = __builtin_amdgcn_wmma_f32_16x16x4_f32(
                  false, a, false, b, (short)0, acc, false, false);
    }

    const int col = n0 + mr;
    const float bv = bias ? bias[col] : 0.f;
    #pragma unroll
    for (int v = 0; v < 8; ++v) {
        const int row = m0 + v + 8 * hi;
        if (row < M) {
            float val = acc[v] + bv;
            if (act) val = elu1(val);
            C[(size_t)row * ncols + col] = val;
        }
    }
}

// ---------------------------------------------------------------------------
// per-(node, head) attention coefficients: alpha_s/d[n,h] = <hW[n,h,:], a_{s/d}[h,:]>
// ---------------------------------------------------------------------------
__global__ void alphas_kernel(const float* __restrict__ hW,
                              const float* __restrict__ att_s,
                              const float* __restrict__ att_d,
                              float* __restrict__ as_, float* __restrict__ ad_,
                              int Nn)
{
    int t = blockIdx.x * blockDim.x + threadIdx.x;
    if (t >= Nn * HEADS) return;
    const int n = t >> 2, hd = t & 3;
    const float* hp = hW + (size_t)n * HIDC + hd * CH;
    const float* sp = att_s + hd * CH;
    const float* dp = att_d + hd * CH;
    float s = 0.f, d = 0.f;
    #pragma unroll
    for (int c = 0; c < CH; ++c) { float h = hp[c]; s += h * sp[c]; d += h * dp[c]; }
    as_[t] = s; ad_[t] = d;
}

// segment max of leaky-relu logits per (dst, head)
__global__ void edge_max_kernel(const int* __restrict__ ei, int E, int Nn,
                                const float* __restrict__ as_,
                                const float* __restrict__ ad_,
                                unsigned* __restrict__ menc)
{
    int e = blockIdx.x * blockDim.x + threadIdx.x;
    if (e >= E + Nn) return;
    int s, d; edge_sd(e, ei, E, s, d);
    #pragma unroll
    for (int hd = 0; hd < HEADS; ++hd) {
        float lg = lrelu(as_[s * 4 + hd] + ad_[d * 4 + hd]);
        atomicMax(&menc[d * 4 + hd], fenc(lg));
    }
}

// segment sum of exp(logit - max) per (dst, head)
__global__ void edge_sum_kernel(const int* __restrict__ ei, int E, int Nn,
                                const float* __restrict__ as_,
                                const float* __restrict__ ad_,
                                const unsigned* __restrict__ menc,
                                float* __restrict__ denom)
{
    int e = blockIdx.x * blockDim.x + threadIdx.x;
    if (e >= E + Nn) return;
    int s, d; edge_sd(e, ei, E, s, d);
    #pragma unroll
    for (int hd = 0; hd < HEADS; ++hd) {
        float lg = lrelu(as_[s * 4 + hd] + ad_[d * 4 + hd]);
        float ex = expf(lg - fdec(menc[d * 4 + hd]));
        atomicAdd(&denom[d * 4 + hd], ex);
    }
}

// weighted message aggregation: out[dst,h,:] += alpha * hW[src,h,:]
// one thread per (edge, head); 32-channel inner loop of f32 atomics.
__global__ void edge_agg_kernel(const int* __restrict__ ei, int E, int Nn,
                                const float* __restrict__ as_,
                                const float* __restrict__ ad_,
                                const unsigned* __restrict__ menc,
                                const float* __restrict__ denom,
                                const float* __restrict__ hW,
                                float* __restrict__ outacc)
{
    long long t = (long long)blockIdx.x * blockDim.x + threadIdx.x;
    if (t >= (long long)(E + Nn) * HEADS) return;
    const int e = (int)(t >> 2), hd = (int)(t & 3);
    int s, d; edge_sd(e, ei, E, s, d);
    float lg = lrelu(as_[s * 4 + hd] + ad_[d * 4 + hd]);
    float w  = expf(lg - fdec(menc[d * 4 + hd])) / denom[d * 4 + hd];
    const float* hp = hW + (size_t)s * HIDC + hd * CH;
    float* op = outacc + (size_t)d * HIDC + hd * CH;
    #pragma unroll
    for (int c = 0; c < CH; ++c) atomicAdd(&op[c], w * hp[c]);
}

// head mean (broadcast over heads) + bias + ELU  -> [N,128]
__global__ void headmean_kernel(const float* __restrict__ acc,
                                const float* __restrict__ bias,
                                float* __restrict__ hout, int Nn)
{
    long long t = (long long)blockIdx.x * blockDim.x + threadIdx.x;
    if (t >= (long long)Nn * HIDC) return;
    const int n = (int)(t >> 7), j = (int)(t & 127), c = j & 31;
    const float* p = acc + (size_t)n * HIDC;
    float m = 0.25f * (p[c] + p[32 + c] + p[64 + c] + p[96 + c]) + bias[j];
    hout[t] = elu1(m);
}

// column sum / sum-of-squares partials (128 threads = 128 columns per block)
__global__ void bnstats_kernel(const float* __restrict__ A, int Nn, int rowsPerBlk,
                               float* __restrict__ colsum, float* __restrict__ colsq)
{
    const int c = threadIdx.x;
    int r0 = blockIdx.x * rowsPerBlk;
    int r1 = r0 + rowsPerBlk; if (r1 > Nn) r1 = Nn;
    float s = 0.f, q = 0.f;
    for (int r = r0; r < r1; ++r) { float v = A[(size_t)r * HIDC + c]; s += v; q += v * v; }
    atomicAdd(&colsum[c], s);
    atomicAdd(&colsq[c], q);
}

__global__ void bnfinalize_kernel(const float* __restrict__ colsum,
                                  const float* __restrict__ colsq,
                                  const float* __restrict__ g,
                                  const float* __restrict__ beta,
                                  float* __restrict__ scale,
                                  float* __restrict__ shift, int Nn)
{
    const int c = threadIdx.x;
    const float inv = 1.f / (float)Nn;
    float mu  = colsum[c] * inv;
    float var = colsq[c] * inv - mu * mu;
    float sc  = g[c] * rsqrtf(var + BN_EPS);
    scale[c] = sc;
    shift[c] = beta[c] - mu * sc;
}

__global__ void bnapply_kernel(float* __restrict__ h,
                               const float* __restrict__ scale,
                               const float* __restrict__ shift, int Nn)
{
    long long t = (long long)blockIdx.x * blockDim.x + threadIdx.x;
    if (t >= (long long)Nn * HIDC) return;
    const int c = (int)(t & 127);
    h[t] = h[t] * scale[c] + shift[c];
}

// final 64 -> 1 projection + sigmoid
__global__ void out_kernel(const float* __restrict__ o1,
                           const float* __restrict__ W2,
                           const float* __restrict__ b2,
                           float* __restrict__ out, int Nn)
{
    int n = blockIdx.x * blockDim.x + threadIdx.x;
    if (n >= Nn) return;
    float acc = b2[0];
    const float* p = o1 + (size_t)n * 64;
    #pragma unroll
    for (int j = 0; j < 64; ++j) acc += p[j] * W2[j];
    out[n] = 1.f / (1.f + expf(-acc));
}

// ---------------------------------------------------------------------------
extern "C" void kernel_launch(void* const* d_in, const int* in_sizes, int n_in,
                              void* d_out, int out_size, void* d_ws, size_t ws_size,
                              hipStream_t stream)
{
    const float* x       = (const float*)d_in[0];
    const int*   ei      = (const int*)  d_in[1];
    const float* W_enc   = (const float*)d_in[2];
    const float* b_enc   = (const float*)d_in[3];
    const float* g_enc   = (const float*)d_in[4];
    const float* be_enc  = (const float*)d_in[5];
    const float* gat_W   = (const float*)d_in[6];
    const float* att_src = (const float*)d_in[7];
    const float* att_dst = (const float*)d_in[8];
    const float* gat_b   = (const float*)d_in[9];
    const float* bn_g    = (const float*)d_in[10];
    const float* bn_b    = (const float*)d_in[11];
    const float* W_skip  = (const float*)d_in[12];
    const float* b_skip  = (const float*)d_in[13];
    const float* W_o1    = (const float*)d_in[14];
    const float* b_o1    = (const float*)d_in[15];
    const float* W_o2    = (const float*)d_in[16];
    const float* b_o2    = (const float*)d_in[17];

    const int Nn = in_sizes[0] / HIDC;     // DIN == HID == 128
    const int E  = in_sizes[1] / 2;
    const int Et = E + Nn;
    const size_t NM = (size_t)Nn * HIDC;

    // workspace layout
    float*    ws      = (float*)d_ws;
    float*    bufA    = ws;                 // current activations [N,128]
    float*    bufB    = ws + NM;            // projected features / z
    float*    bufC    = ws + 2 * NM;        // aggregation accumulator / o1
    float*    alpha_s = ws + 3 * NM;        // [N,4]
    float*    alpha_d = alpha_s + (size_t)Nn * HEADS;
    unsigned* menc    = (unsigned*)(alpha_d + (size_t)Nn * HEADS);
    float*    denom   = (float*)(menc + (size_t)Nn * HEADS);
    float*    colsum  = denom + (size_t)Nn * HEADS;   // [128]
    float*    colsq   = colsum + HIDC;                // [128]
    float*    bnscale = colsq + HIDC;                 // [128]
    float*    bnshift = bnscale + HIDC;               // [128]

    const int MB = (Nn + 15) / 16;          // 16-row blocks (3125)
    const int ROWS_PB = 256;

    auto runBN = [&](float* buf, const float* g, const float* beta) {
        hipMemsetAsync(colsum, 0, 2 * HIDC * sizeof(float), stream);
        bnstats_kernel<<<(Nn + ROWS_PB - 1) / ROWS_PB, HIDC, 0, stream>>>(
            buf, Nn, ROWS_PB, colsum, colsq);
        bnfinalize_kernel<<<1, HIDC, 0, stream>>>(colsum, colsq, g, beta,
                                                  bnscale, bnshift, Nn);
        bnapply_kernel<<<(int)((NM + 255) / 256), 256, 0, stream>>>(
            buf, bnscale, bnshift, Nn);
    };

    // encoder: elu(x @ W_enc + b_enc), then BN
    wmma_gemm_kernel<<<MB, 256, 0, stream>>>(x, nullptr, W_enc, b_enc,
                                             bufA, Nn, 128, 128, 1);
    runBN(bufA, g_enc, be_enc);

    for (int l = 0; l < 3; ++l) {
        // projection (no bias / activation)
        wmma_gemm_kernel<<<MB, 256, 0, stream>>>(bufA, nullptr,
                                                 gat_W + (size_t)l * HIDC * HIDC,
                                                 nullptr, bufB, Nn, 128, 128, 0);
        alphas_kernel<<<(Nn * HEADS + 255) / 256, 256, 0, stream>>>(
            bufB, att_src + l * HIDC, att_dst + l * HIDC, alpha_s, alpha_d, Nn);

        hipMemsetAsync(menc,  0, (size_t)Nn * HEADS * sizeof(unsigned), stream);
        hipMemsetAsync(denom, 0, (size_t)Nn * HEADS * sizeof(float), stream);
        hipMemsetAsync(bufC,  0, NM * sizeof(float), stream);

        edge_max_kernel<<<(Et + 255) / 256, 256, 0, stream>>>(
            ei, E, Nn, alpha_s, alpha_d, menc);
        edge_sum_kernel<<<(Et + 255) / 256, 256, 0, stream>>>(
            ei, E, Nn, alpha_s, alpha_d, menc, denom);
        edge_agg_kernel<<<(int)(((long long)Et * HEADS + 255) / 256), 256, 0, stream>>>(
            ei, E, Nn, alpha_s, alpha_d, menc, denom, bufB, bufC);

        headmean_kernel<<<(int)((NM + 255) / 256), 256, 0, stream>>>(
            bufC, gat_b + l * HIDC, bufA, Nn);
        runBN(bufA, bn_g + l * HIDC, bn_b + l * HIDC);
    }

    // skip: elu(concat([h, x]) @ W_skip + b_skip)   (K = 256 via A/A2)
    wmma_gemm_kernel<<<MB, 256, 0, stream>>>(bufA, x, W_skip, b_skip,
                                             bufB, Nn, 256, 128, 1);
    // o1: elu(z @ W_o1 + b_o1)   (64 output cols -> 4 waves)
    wmma_gemm_kernel<<<MB, 128, 0, stream>>>(bufB, nullptr, W_o1, b_o1,
                                             bufC, Nn, 128, 64, 1);
    // out: sigmoid(o1 @ W_o2 + b_o2)
    out_kernel<<<(Nn + 255) / 256, 256, 0, stream>>>(bufC, W_o2, b_o2,
                                                     (float*)d_out, Nn);
}